// GGNNMsgPass_73323681677623
// MI455X (gfx1250) — compile-verified
//
#include <hip/hip_runtime.h>
#include <stdint.h>

typedef __attribute__((ext_vector_type(2))) float v2f;
typedef __attribute__((ext_vector_type(8))) float v8f;
typedef __attribute__((ext_vector_type(4))) unsigned int v4u;
typedef __attribute__((ext_vector_type(8))) int v8i;
typedef __attribute__((ext_vector_type(4))) int v4i;

#define NN 1024
#define DD 64
#define CHUNK_ROWS 64
#define NCHUNK (NN / CHUNK_ROWS)     // 16 chunks
#define LDS_ROW 80                   // 64 data dwords + 16 pad dwords (TDM padding)
#define BUF_FLTS (CHUNK_ROWS * LDS_ROW)  // 5120 floats = 20 KB per buffer
#define AGG_ROW 66                   // padded agg row -> conflict-free transposed read

// TDM: load CHUNK_ROWS x 64 f32 tile of feat into LDS, padding rows 64dw -> 80dw
__device__ __forceinline__ void tdm_feat_load(const float* gptr, uint32_t lds_byte_addr) {
    uint64_t ga = (uint64_t)(uintptr_t)gptr;
    v4u g0;
    g0.x = 1u;                                               // count=1 valid user descriptor
    g0.y = lds_byte_addr;                                    // lds_addr (bytes)
    g0.z = (uint32_t)(ga & 0xffffffffu);                     // global_addr lo
    g0.w = (uint32_t)((ga >> 32) & 0x01ffffffu) | (2u << 30);// global_addr hi | type=2
    v8i g1;
    g1[0] = (int)((2u << 16) | (1u << 20) | (5u << 22) | (15u << 25)); // 4B elems, pad 16dw every 64dw
    g1[1] = (int)(64u << 16);     // tensor_dim0 = 64
    g1[2] = (int)(1024u << 16);   // tensor_dim1 = 1024
    g1[3] = (int)(64u << 16);     // tile_dim0 = 64
    g1[4] = CHUNK_ROWS;           // tile_dim1 = 64
    g1[5] = 64;                   // tensor_dim0_stride = 64
    g1[6] = 0;
    g1[7] = 0;
    v4i gz = {};
    v8i gz8 = {};
    __builtin_amdgcn_tensor_load_to_lds(g0, g1, gz, gz, gz8, 0);
}

__global__ __launch_bounds__(256) void ggnn_msgpass_kernel(
    const float* __restrict__ feat, const int* __restrict__ adj,
    const float* __restrict__ eW_in, const float* __restrict__ eW_out,
    const float* __restrict__ bias, float* __restrict__ out)
{
    // 40 KB: 2 x (64 rows x 80 dwords) feat buffers, reused for agg tiles (8*16 x 66 = 33 KB)
    __shared__ float lds[2 * BUF_FLTS];

    const int bb   = blockIdx.x >> 6;          // batch
    const int i0   = (blockIdx.x & 63) << 4;   // node-row tile base
    const int wave = threadIdx.x >> 5;
    const int lane = threadIdx.x & 31;
    const int lo   = lane & 15;
    const int hi   = (lane >> 4) << 1;         // 0 or 2 : K offset of half-wave
    const int h8   = (lane >> 4) << 3;         // 0 or 8 : M offset in C/D tile

    const int dir  = wave >> 2;                // 0 = in-edges (adj), 1 = out-edges (adj^T)
    const int cls  = wave & 3;                 // edge class owned by this wave
    const int cval = cls + 1;                  // adj code for this class

    const int* adjB = adj + (size_t)bb * NN * NN;
    const float* featB = feat + (size_t)bb * NN * DD;

    v8f acc0 = {}, acc1 = {}, acc2 = {}, acc3 = {};   // 16x16 agg tiles, e-chunks 0..3

    const int* inP   = adjB + (i0 + lo) * NN + hi;    // + jj      (rows of the i-tile)
    const int* outP0 = adjB + hi * NN + (i0 + lo);    // + jj*NN   (adj^T rows)

    // prefetch chunk 0 into buffer 0
    if (threadIdx.x == 0)
        tdm_feat_load(featB, (uint32_t)(uintptr_t)(void*)lds);

    for (int ch = 0; ch < NCHUNK; ++ch) {
        __syncthreads();   // everyone done with buffer (ch+1)&1 from chunk ch-1
        if (threadIdx.x == 0) {
            if (ch + 1 < NCHUNK) {
                tdm_feat_load(featB + (size_t)(ch + 1) * CHUNK_ROWS * DD,
                              (uint32_t)(uintptr_t)(void*)(lds + ((ch + 1) & 1) * BUF_FLTS));
                __builtin_amdgcn_s_wait_tensorcnt(1);   // in-order: chunk ch has landed
            } else {
                __builtin_amdgcn_s_wait_tensorcnt(0);
            }
        }
        __syncthreads();

        const int jbase = ch * CHUNK_ROWS;
        const float* buf = lds + (ch & 1) * BUF_FLTS;
#pragma unroll 4
        for (int t = 0; t < CHUNK_ROWS / 4; ++t) {   // K=4 per step
            const int jj = jbase + t * 4;
            int a0, a1;
            if (dir == 0) {
                const int2 av = *(const int2*)(inP + jj);   // adj[b, i0+lo, jj+hi .. +1]
                a0 = av.x; a1 = av.y;
            } else {
                const int* p = outP0 + jj * NN;             // adj[b, jj+hi(+1), i0+lo]
                a0 = p[0];
                a1 = p[NN];
            }
            v2f m;                                          // A: one class mask, reused 4x
            m.x = (a0 == cval) ? 1.0f : 0.0f;
            m.y = (a1 == cval) ? 1.0f : 0.0f;

            const float* L = buf + (t * 4 + hi) * LDS_ROW + lo;  // B: feat rows jj+K
            v2f b0, b1, b2, b3;
            b0.x = L[0];  b0.y = L[LDS_ROW];
            b1.x = L[16]; b1.y = L[LDS_ROW + 16];
            b2.x = L[32]; b2.y = L[LDS_ROW + 32];
            b3.x = L[48]; b3.y = L[LDS_ROW + 48];

            acc0 = __builtin_amdgcn_wmma_f32_16x16x4_f32(false, m, false, b0, (short)0, acc0, false, false);
            acc1 = __builtin_amdgcn_wmma_f32_16x16x4_f32(false, m, false, b1, (short)0, acc1, false, false);
            acc2 = __builtin_amdgcn_wmma_f32_16x16x4_f32(false, m, false, b2, (short)0, acc2, false, false);
            acc3 = __builtin_amdgcn_wmma_f32_16x16x4_f32(false, m, false, b3, (short)0, acc3, false, false);
        }
    }

    // Phase 2: spill agg tiles (C/D layout: M = r+h8, N = lo) to LDS, padded rows
    __syncthreads();
    {
        float* aggB = lds + ((dir * 4 + cls) * 16) * AGG_ROW;
#pragma unroll
        for (int r = 0; r < 8; ++r) {
            const int row = r + h8;
            aggB[row * AGG_ROW +  0 + lo] = acc0[r];
            aggB[row * AGG_ROW + 16 + lo] = acc1[r];
            aggB[row * AGG_ROW + 32 + lo] = acc2[r];
            aggB[row * AGG_ROW + 48 + lo] = acc3[r];
        }
    }
    __syncthreads();

    // Phase 3: msg[i,d] = sum_c sum_e agg[dir][c][i][e] * eW[c][d][e]  (wave = (dir, d-chunk))
    {
        const int dc = cls;                    // reuse wave split as d-chunk
        const float* eW = (dir == 0) ? eW_in : eW_out;
        v8f msg = {};
#pragma unroll
        for (int c = 0; c < 4; ++c) {
            const float* eWc    = eW + c * DD * DD + (dc * 16 + lo) * DD;    // row d = dc*16+lo
            const float* aggRow = &lds[((dir * 4 + c) * 16 + lo) * AGG_ROW]; // row i = lo
#pragma unroll
            for (int ks = 0; ks < 16; ++ks) {
                const int e = ks * 4 + hi;
                v2f a, b;
                a.x = aggRow[e]; a.y = aggRow[e + 1];   // A: agg[i=lo][K=e]
                b.x = eWc[e];    b.y = eWc[e + 1];      // B: eW[c][d][K=e]
                msg = __builtin_amdgcn_wmma_f32_16x16x4_f32(false, a, false, b, (short)0, msg, false, false);
            }
        }
        const float badd = bias[dir * DD + dc * 16 + lo];
        float* op = out + ((size_t)(bb * NN + i0 + h8)) * 128 + dir * DD + dc * 16 + lo;
#pragma unroll
        for (int r = 0; r < 8; ++r)
            op[(size_t)r * 128] = msg[r] + badd;
    }
}

extern "C" void kernel_launch(void* const* d_in, const int* in_sizes, int n_in,
                              void* d_out, int out_size, void* d_ws, size_t ws_size,
                              hipStream_t stream) {
    (void)in_sizes; (void)n_in; (void)out_size; (void)d_ws; (void)ws_size;
    const float* feat   = (const float*)d_in[0];
    const int*   adj    = (const int*)d_in[1];
    const float* eW_in  = (const float*)d_in[2];
    const float* eW_out = (const float*)d_in[3];
    const float* bias   = (const float*)d_in[4];
    float*       out    = (float*)d_out;
    // one block per (batch, 16-row node tile): 16 * 64 = 1024 blocks, 8 waves each
    ggnn_msgpass_kernel<<<dim3(16 * 64), dim3(256), 0, stream>>>(feat, adj, eW_in, eW_out, bias, out);
}